// CSTDecoder18_10660108828907
// MI455X (gfx1250) — compile-verified
//
#include <hip/hip_runtime.h>

typedef __attribute__((ext_vector_type(4))) float f4;
typedef __attribute__((ext_vector_type(2))) float f2;

#define ZC    18      // scalars per row in z
#define NXS   2048    // x samples per row
#define TPB   256     // threads per block (8 wave32)
#define EPSV  1e-8f

// Forced VOP3P packed-f32 math (CDNA5 V_PK_MUL_F32 / V_PK_FMA_F32).
__device__ __forceinline__ f2 pk_mul(f2 a, f2 b) {
    f2 d;
    asm("v_pk_mul_f32 %0, %1, %2" : "=v"(d) : "v"(a), "v"(b));
    return d;
}
__device__ __forceinline__ f2 pk_fma(f2 a, f2 b, f2 c) {
    f2 d;
    asm("v_pk_fma_f32 %0, %1, %2, %3" : "=v"(d) : "v"(a), "v"(b), "v"(c));
    return d;
}

__device__ __forceinline__ f2 clamp01(f2 a) {
    return __builtin_elementwise_min(__builtin_elementwise_max(a, (f2)0.f), (f2)1.f);
}

// Uniform-surface Horner: P = sum_k cw[k] * x^k * u^(7-k); cw2 pre-splatted
// f2 pairs -> the whole chain is pair-flowing pk math, no marshalling.
__device__ __forceinline__ f2 horner_uni(const f2* cw2, f2 xx, f2 u) {
    f2 P   = cw2[7];
    f2 upw = u;
#pragma unroll
    for (int k = 6; k >= 0; --k) {
        P = pk_fma(P, xx, pk_mul(cw2[k], upw));
        if (k) upw = pk_mul(upw, u);
    }
    return P;
}

// Mixed-surface Horner (chunk straddles i_le; rare): exact per-element selects.
__device__ __forceinline__ f2 horner_mix(const float* cwl, const float* cwu,
                                         bool up0, bool up1, f2 xx, f2 u) {
    f2 c;
    c[0] = up0 ? cwu[7] : cwl[7];
    c[1] = up1 ? cwu[7] : cwl[7];
    f2 P   = c;
    f2 upw = u;
#pragma unroll
    for (int k = 6; k >= 0; --k) {
        c[0] = up0 ? cwu[k] : cwl[k];
        c[1] = up1 ? cwu[k] : cwl[k];
        P = pk_fma(P, xx, pk_mul(c, upw));
        if (k) upw = pk_mul(upw, u);
    }
    return P;
}

// Shared tail: class function + leading-edge term (identical for both surfaces).
__device__ __forceinline__ f2 cst_tail(f2 xx, f2 u, f2 P, f2 lew2) {
    f2 xc = __builtin_elementwise_min(
                __builtin_elementwise_max(xx, (f2)EPSV), (f2)(1.f - EPSV));
    f2 C = pk_mul(__builtin_elementwise_sqrt(xc), 1.f - xc);
    f2 y = pk_mul(C, P);
    f2 v2 = pk_mul(u, u);
    f2 v4 = pk_mul(v2, v2);
    f2 v8 = pk_mul(v4, v4);
    f2 le = pk_mul(v8, __builtin_elementwise_sqrt(u));
    f2 lx = pk_mul(xx, lew2);
    return pk_fma(lx, le, y);
}

// One 4-element chunk starting at column n: branch picks only P and TE sign;
// clamp/u/te products and the tail are shared across paths.
__device__ __forceinline__ void cst_chunk(f4 v, int n, int i_le,
                                          const f2* cw2l, const f2* cw2u,
                                          const float* cwl, const float* cwu,
                                          f2 lew2, f2 te2,
                                          f2& ya, f2& yb)
{
    f2 pa = {v[0], v[1]};
    f2 pb = {v[2], v[3]};
    f2 xxa = clamp01(pa),  xxb = clamp01(pb);
    f2 ua  = 1.f - xxa,    ub  = 1.f - xxb;
    f2 tea = pk_mul(xxa, te2), teb = pk_mul(xxb, te2);

    f2 Pa, Pb, ta, tb;
    if (n + 3 <= i_le) {                 // whole chunk upper surface
        Pa = horner_uni(cw2u, xxa, ua);
        Pb = horner_uni(cw2u, xxb, ub);
        ta = tea;  tb = teb;
    } else if (n > i_le) {               // whole chunk lower surface
        Pa = horner_uni(cw2l, xxa, ua);
        Pb = horner_uni(cw2l, xxb, ub);
        ta = -tea; tb = -teb;
    } else {                             // straddles the LE point (rare)
        const bool u0 = (n     <= i_le);
        const bool u1 = (n + 1 <= i_le);
        const bool u2 = (n + 2 <= i_le);
        const bool u3 = (n + 3 <= i_le);
        Pa = horner_mix(cwl, cwu, u0, u1, xxa, ua);
        Pb = horner_mix(cwl, cwu, u2, u3, xxb, ub);
        ta[0] = u0 ? tea[0] : -tea[0];
        ta[1] = u1 ? tea[1] : -tea[1];
        tb[0] = u2 ? teb[0] : -teb[0];
        tb[1] = u3 ? teb[1] : -teb[1];
    }

    ya = cst_tail(xxa, ua, Pa, lew2) + ta;
    yb = cst_tail(xxb, ub, Pb, lew2) + tb;
}

__global__ __launch_bounds__(TPB, 2)
void cst_decode(const float* __restrict__ z,
                const float* __restrict__ x,
                float* __restrict__ out)
{
    const int b   = blockIdx.x;
    const int tid = threadIdx.x;

    __shared__ unsigned long long wavemin[TPB / 32];
    __shared__ unsigned long long rowkey;

    // Row scalars: uniform address -> scalar loads (s_load) into SGPRs.
    const float* zr = z + (size_t)b * ZC;
    const float binom[8] = {1.f, 7.f, 21.f, 35.f, 35.f, 21.f, 7.f, 1.f};
    float cwl[8], cwu[8];
    f2 cw2l[8], cw2u[8];
#pragma unroll
    for (int k = 0; k < 8; ++k) {
        cwl[k] = binom[k] * zr[k];
        cwu[k] = binom[k] * zr[8 + k];
        cw2l[k] = (f2)cwl[k];
        cw2u[k] = (f2)cwu[k];
    }
    const f2 lew2 = (f2)zr[16];
    const f2 te2  = (f2)(zr[17] * 0.5f);

    // Vector loads: elements [4*tid .. 4*tid+3] and [1024+4*tid .. +3].
    const f4* __restrict__ xrow = (const f4*)(x + (size_t)b * NXS);
    const f4 v0 = xrow[tid];
    const f4 v1 = xrow[TPB + tid];
    const int n0 = tid * 4;
    const int n1 = 4 * TPB + tid * 4;

    // ---- Phase 1: row argmin (first occurrence) via packed (bits<<32)|index.
    unsigned long long key = ~0ull;
#pragma unroll
    for (int i = 0; i < 4; ++i) {
        float c0 = fminf(fmaxf(v0[i], 0.f), 1.f);
        float c1 = fminf(fmaxf(v1[i], 0.f), 1.f);
        unsigned long long k0 =
            ((unsigned long long)__float_as_uint(c0) << 32) | (unsigned)(n0 + i);
        unsigned long long k1 =
            ((unsigned long long)__float_as_uint(c1) << 32) | (unsigned)(n1 + i);
        key = (k0 < key) ? k0 : key;   // -> v_min_u64
        key = (k1 < key) ? k1 : key;
    }
#pragma unroll
    for (int off = 16; off > 0; off >>= 1) {
        unsigned long long o = __shfl_xor(key, off, 32);
        key = (o < key) ? o : key;
    }
    if ((tid & 31) == 0) wavemin[tid >> 5] = key;
    __syncthreads();
    if (tid == 0) {
        unsigned long long m = wavemin[0];
#pragma unroll
        for (int w = 1; w < TPB / 32; ++w) m = (wavemin[w] < m) ? wavemin[w] : m;
        rowkey = m;
    }
    __syncthreads();
    const int i_le = (int)(unsigned)(rowkey & 0xFFFFFFFFu);

    // ---- Phase 2: evaluate y, write interleaved [x_orig, y] NT b128.
    f4* __restrict__ orow = (f4*)(out + (size_t)b * (2 * NXS));

    {
        f2 ya, yb;
        cst_chunk(v0, n0, i_le, cw2l, cw2u, cwl, cwu, lew2, te2, ya, yb);
        f4 o0 = {v0[0], ya[0], v0[1], ya[1]};
        f4 o1 = {v0[2], yb[0], v0[3], yb[1]};
        __builtin_nontemporal_store(o0, &orow[tid * 2 + 0]);
        __builtin_nontemporal_store(o1, &orow[tid * 2 + 1]);
    }
    {
        f2 ya, yb;
        cst_chunk(v1, n1, i_le, cw2l, cw2u, cwl, cwu, lew2, te2, ya, yb);
        f4 o0 = {v1[0], ya[0], v1[1], ya[1]};
        f4 o1 = {v1[2], yb[0], v1[3], yb[1]};
        __builtin_nontemporal_store(o0, &orow[2 * TPB + tid * 2 + 0]);
        __builtin_nontemporal_store(o1, &orow[2 * TPB + tid * 2 + 1]);
    }
}

extern "C" void kernel_launch(void* const* d_in, const int* in_sizes, int n_in,
                              void* d_out, int out_size, void* d_ws, size_t ws_size,
                              hipStream_t stream)
{
    const float* z = (const float*)d_in[0];   // (B, 18) f32
    const float* x = (const float*)d_in[1];   // (B, 2048) f32
    float* out     = (float*)d_out;           // (B, 4096) f32

    const int B = in_sizes[0] / ZC;           // 4096
    cst_decode<<<dim3(B), dim3(TPB), 0, stream>>>(z, x, out);
}